// TextSemanticEncoding_26474178412866
// MI455X (gfx1250) — compile-verified
//
#include <hip/hip_runtime.h>
#include <hip/hip_bf16.h>
#include <math.h>

typedef __attribute__((ext_vector_type(2))) float v2f;
typedef __attribute__((ext_vector_type(8))) float v8f;

#define S_LEN   200
#define H_DIM   128
#define CHUNK   40            // rows of emb per LDS chunk (S_LEN = 5 * CHUNK)
#define NCHUNK  5
#define ROW_F   132           // padded LDS row stride in floats (528 B, kills bank conflicts)
#define WAVES   8             // 256 threads / wave32

// ---- feature detection (device pass only) ----------------------------------
#if defined(__HIP_DEVICE_COMPILE__) && __has_builtin(__builtin_amdgcn_global_load_async_to_lds_b128)
#define USE_ASYNC 1
#else
#define USE_ASYNC 0
#endif

#if defined(__HIP_DEVICE_COMPILE__) && __has_builtin(__builtin_amdgcn_wmma_f32_16x16x4_f32)
#define HAVE_WMMA 1
#else
#define HAVE_WMMA 0
#endif

#if USE_ASYNC
// Builtin signature (from hipcc diagnostic): param 0 is
// 'int __attribute__((vector_size(16)))' pointer in global (AS1) space.
typedef int v4i_gcc __attribute__((vector_size(16)));
typedef __attribute__((address_space(1))) v4i_gcc as1_v4i;
typedef __attribute__((address_space(3))) v4i_gcc as3_v4i;

__device__ __forceinline__ void async_copy16(const float* gp, float* lp) {
  // copies 16 bytes per lane, global -> LDS, tracked by ASYNCcnt
  __builtin_amdgcn_global_load_async_to_lds_b128((as1_v4i*)gp, (as3_v4i*)lp, 0, 0);
}

__device__ __forceinline__ void wait_async0() {
#if __has_builtin(__builtin_amdgcn_s_wait_asynccnt)
  __builtin_amdgcn_s_wait_asynccnt(0);
#else
  asm volatile("s_wait_asynccnt 0" ::: "memory");
#endif
}
#endif

// ----------------------------------------------------------------------------
// One block per batch b. 8 waves; wave w owns h-columns [16w, 16w+16).
// Global->LDS double-buffered async pipeline; WMMA f32 16x16x4 reduction.
// ----------------------------------------------------------------------------
__global__ __launch_bounds__(256)
void ctx_pool_kernel(const float* __restrict__ emb,
                     const int*  __restrict__ lengths,
                     const int*  __restrict__ gnum_p,
                     float*      __restrict__ out)
{
  __shared__ __align__(16) float ws[S_LEN];               // per-batch decay weights
  __shared__ __align__(16) float buf[2][CHUNK * ROW_F];   // double-buffered emb chunks

  const int b    = blockIdx.x;
  const int tid  = threadIdx.x;
  const int lane = tid & 31;
  const int wv   = tid >> 5;

  // ---- per-batch decay weights (faithful to reference _decay_mask) ----
  if (tid < S_LEN) {
    const int G = gnum_p[0];
    const int L = lengths[b];
    float w = 1.0f;
    if (L >= G && tid < L) {
      int gs = L / G; if (gs < 1) gs = 1;
      int g  = tid / gs; if (g > G - 1) g = G - 1;
      int ge = (g == G - 1) ? L : (g + 1) * gs;
      w = __powf(0.8f, (float)(ge - tid - 1)) * __powf(0.6f, (float)(G - 1 - g));
    }
    ws[tid] = w;
  }

  const float* embb = emb + (size_t)b * (S_LEN * H_DIM);

  // ---- prefetch chunk 0: wave wv moves rows wv, wv+8, ... (512 B / instr) ----
  for (int j = 0; j < CHUNK / WAVES; ++j) {
    const int row = wv + WAVES * j;
    const float* gp = embb + row * H_DIM + lane * 4;
    float*       lp = &buf[0][row * ROW_F] + lane * 4;
#if USE_ASYNC
    async_copy16(gp, lp);
#else
    *(float4*)lp = *(const float4*)gp;
#endif
  }

  const int h0 = wv * 16;
  const int q  = lane >> 4;   // K half: lanes 0-15 -> K{0,1}, lanes 16-31 -> K{2,3}
  const int n  = lane & 15;   // N (h column within tile)
  v8f acc = {};

  for (int c = 0; c < NCHUNK; ++c) {
#if USE_ASYNC
    wait_async0();            // my chunk-c rows are in LDS
#endif
    __syncthreads();          // everyone's chunk-c rows visible; prior readers of next buf done

    if (c + 1 < NCHUNK) {     // kick off next chunk into the other buffer
      const int nb = (c + 1) & 1;
      for (int j = 0; j < CHUNK / WAVES; ++j) {
        const int row = wv + WAVES * j;
        const float* gp = embb + ((c + 1) * CHUNK + row) * H_DIM + lane * 4;
        float*       lp = &buf[nb][row * ROW_F] + lane * 4;
#if USE_ASYNC
        async_copy16(gp, lp);
#else
        *(float4*)lp = *(const float4*)gp;
#endif
      }
    }

    // ---- consume chunk c: 10 WMMA steps of K=4 rows each ----
    const float* cb = buf[c & 1];
    const float* wc = ws + c * CHUNK;
    #pragma unroll
    for (int r = 0; r < CHUNK; r += 4) {
      const int sr = r + 2 * q;                 // this lane-group's first K row
      v2f av = *(const v2f*)(wc + sr);          // A: w[s+k] broadcast over M
      v2f bv;
      bv[0] = cb[(sr + 0) * ROW_F + h0 + n];    // B: emb[s+k, h0+n]
      bv[1] = cb[(sr + 1) * ROW_F + h0 + n];
#if HAVE_WMMA
      acc = __builtin_amdgcn_wmma_f32_16x16x4_f32(
          /*neg_a=*/false, av, /*neg_b=*/false, bv,
          /*c_mod=*/(short)0, acc, /*reuse_a=*/false, /*reuse_b=*/false);
#else
      acc[0] += av[0] * bv[0] + av[1] * bv[1];
#endif
    }
  }

  // ---- writeback: D row 0 lives in acc[0], lanes 0-15 (all M rows identical) ----
#if HAVE_WMMA
  const float res = acc[0];
#else
  const float res = acc[0] + __shfl(acc[0], lane ^ 16, 32);
#endif
  if (lane < 16) {
    out[(size_t)b * H_DIM + h0 + lane] = res;
  }
}

extern "C" void kernel_launch(void* const* d_in, const int* in_sizes, int n_in,
                              void* d_out, int out_size, void* d_ws, size_t ws_size,
                              hipStream_t stream) {
  const float* emb     = (const float*)d_in[0];   // [B, 200, 128] f32
  const int*   lengths = (const int*)d_in[1];     // [B] i32
  const int*   gnum    = (const int*)d_in[2];     // scalar i32 (=4)
  float*       out     = (float*)d_out;           // [B, 128] f32
  (void)n_in; (void)out_size; (void)d_ws; (void)ws_size;

  const int B = in_sizes[1];                      // 4096 batches
  ctx_pool_kernel<<<B, 256, 0, stream>>>(emb, lengths, gnum, out);
}